// GCN_Policy_Module_86053964742745
// MI455X (gfx1250) — compile-verified
//
#include <hip/hip_runtime.h>
#include <math.h>

typedef __attribute__((ext_vector_type(2))) float v2f;
typedef __attribute__((ext_vector_type(8))) float v8f;

#define SLOPE 0.01f

// ---------------------------------------------------------------- utilities
__global__ void zero_f32(float* __restrict__ p, int n) {
  int i = blockIdx.x * blockDim.x + threadIdx.x;
  if (i < n) p[i] = 0.0f;
}

__global__ void deg_count(const int* __restrict__ dst, float* __restrict__ deg, int E) {
  int e = blockIdx.x * blockDim.x + threadIdx.x;
  if (e < E) atomicAdd(&deg[dst[e]], 1.0f);
}

__global__ void finalize_dinv(float* __restrict__ dinv, int n) {
  int i = blockIdx.x * blockDim.x + threadIdx.x;
  if (i < n) dinv[i] = rsqrtf(dinv[i] + 1.0f);
}

// ---------------------------------------------------------------- W prepack
// Swizzle W[K x 32] into per-lane B-fragment order so each lane fetches one
// float4 (b128) per k-step: Wsw[s*128 + lane*4 + {b0.x,b0.y,b1.x,b1.y}]
// where b0 covers cols 0..15 and b1 covers cols 16..31, K rows (4s+2hi, +1).
__global__ void prepack_w(const float* __restrict__ W, float* __restrict__ Wsw, int K) {
  int t = blockIdx.x * blockDim.x + threadIdx.x;
  if (t >= K * 32) return;
  int s   = t >> 7;          // k-step (4 K-rows per step)
  int rem = t & 127;
  int l   = rem >> 2;        // lane 0..31
  int c   = rem & 3;         // component
  int hi = l >> 4, lo = l & 15;
  int k = 4 * s + 2 * hi + (c & 1);
  int n = lo + ((c & 2) ? 16 : 0);
  Wsw[t] = W[k * 32 + n];
}

// ---------------------------------------------------------------- WMMA GEMM
// H[nrows x 32] = X[nrows x K] @ W[K x 32], fp32, V_WMMA_F32_16X16X4_F32.
// One wave per 16-row tile; both 16-col halves accumulated in a single K loop
// so the layer-1 call may run in place (Hout == X). Assumes nrows % 16 == 0
// (N = 100000 = 6250 * 16), so no per-lane bounds guards in the hot loop.
__global__ __launch_bounds__(128)
void gemm_wmma32(const float* __restrict__ X, const float* __restrict__ Wsw,
                 float* __restrict__ Hout, int nrows, int K) {
  const int lane = threadIdx.x & 31;
  const int wave = threadIdx.x >> 5;
  const int tile = blockIdx.x * 4 + wave;
  const int row0 = tile * 16;
  if (row0 >= nrows) return;               // whole wave exits: EXEC full for WMMA

  const int hi = lane >> 4;                // 0: K pair {0,1}, 1: K pair {2,3}
  const int lo = lane & 15;

  const float2* __restrict__ Xp = (const float2*)(X + (size_t)(row0 + lo) * K);
  const float4* __restrict__ Bp = (const float4*)Wsw;

  v8f acc0 = {};                           // cols 0..15
  v8f acc1 = {};                           // cols 16..31

  const int steps = K >> 2;
  for (int s = 0; s < steps; ++s) {
    float2 af = Xp[s * 2 + hi];            // global_load_b64
    float4 bf = Bp[s * 32 + lane];         // global_load_b128, coalesced
    v2f a;   a.x   = af.x; a.y   = af.y;
    v2f b0v; b0v.x = bf.x; b0v.y = bf.y;
    v2f b1v; b1v.x = bf.z; b1v.y = bf.w;
    acc0 = __builtin_amdgcn_wmma_f32_16x16x4_f32(false, a, false, b0v,
                                                 (short)0, acc0, false, false);
    acc1 = __builtin_amdgcn_wmma_f32_16x16x4_f32(false, a, false, b1v,
                                                 (short)0, acc1, false, false);
  }

  // C/D layout: VGPR j holds row (row0 + hi*8 + j), column lo (+16 for acc1)
#pragma unroll
  for (int j = 0; j < 8; ++j) {
    int r = row0 + hi * 8 + j;
    Hout[(size_t)r * 32 + lo]      = acc0[j];
    Hout[(size_t)r * 32 + lo + 16] = acc1[j];
  }
}

// ---------------------------------------------------------------- edge scatter
// 8 threads per edge, each moves a float4 slice: coalesced 128B gather per edge.
__global__ void scatter32(const float* __restrict__ h, const int* __restrict__ src,
                          const int* __restrict__ dst, const float* __restrict__ dinv,
                          float* __restrict__ agg, int E) {
  int t = blockIdx.x * blockDim.x + threadIdx.x;
  int e = t >> 3;
  int q = t & 7;
  if (e >= E) return;
  int s = src[e], d = dst[e];
  float nrm = dinv[s] * dinv[d];
  const float4 v = ((const float4*)(h + (size_t)s * 32))[q];
  float* ap = agg + (size_t)d * 32 + q * 4;
  atomicAdd(ap + 0, v.x * nrm);
  atomicAdd(ap + 1, v.y * nrm);
  atomicAdd(ap + 2, v.z * nrm);
  atomicAdd(ap + 3, v.w * nrm);
}

__global__ void scatter1(const float* __restrict__ lp, const int* __restrict__ src,
                         const int* __restrict__ dst, const float* __restrict__ dinv,
                         float* __restrict__ agg2, int E) {
  int e = blockIdx.x * blockDim.x + threadIdx.x;
  if (e >= E) return;
  int s = src[e], d = dst[e];
  atomicAdd(&agg2[d], lp[s] * dinv[s] * dinv[d]);
}

// ---------------------------------------------------------------- combine
// h[n,f] = LeakyReLU(agg[n,f] + h[n,f]*dinv[n]^2 + b[f]) written in place.
__global__ void combine32(float* __restrict__ h, const float* __restrict__ agg,
                          const float* __restrict__ dinv, const float* __restrict__ b,
                          int n) {
  int t = blockIdx.x * blockDim.x + threadIdx.x;
  if (t >= n * 32) return;
  int node = t >> 5, f = t & 31;
  float sn = dinv[node]; sn *= sn;
  float v = agg[t] + h[t] * sn + b[f];
  h[t] = (v >= 0.0f) ? v : SLOPE * v;
}

// lp[n] = h[n,:] . W2 (32 -> 1 projection)
__global__ void dot32(const float* __restrict__ h, const float* __restrict__ W2,
                      float* __restrict__ lp, int n) {
  int i = blockIdx.x * blockDim.x + threadIdx.x;
  if (i >= n) return;
  const float4* hp = (const float4*)(h + (size_t)i * 32);
  const float4* wp = (const float4*)W2;
  float s = 0.0f;
#pragma unroll
  for (int q = 0; q < 8; ++q) {
    float4 a = hp[q], w = wp[q];
    s += a.x * w.x + a.y * w.y + a.z * w.z + a.w * w.w;
  }
  lp[i] = s;
}

// logits = agg2 + lp*dinv^2 + b2 ; out = sigmoid(logits). d_out = [out | logits]
__global__ void final_head(const float* __restrict__ lp, const float* __restrict__ agg2,
                           const float* __restrict__ dinv, const float* __restrict__ b2,
                           float* __restrict__ out, int n) {
  int i = blockIdx.x * blockDim.x + threadIdx.x;
  if (i >= n) return;
  float sn = dinv[i]; sn *= sn;
  float logit = agg2[i] + lp[i] * sn + b2[0];
  out[i]     = 1.0f / (1.0f + expf(-logit));
  out[n + i] = logit;
}

// ---------------------------------------------------------------- launcher
extern "C" void kernel_launch(void* const* d_in, const int* in_sizes, int n_in,
                              void* d_out, int out_size, void* d_ws, size_t ws_size,
                              hipStream_t stream) {
  const float* x  = (const float*)d_in[0];
  const int* eidx = (const int*)d_in[1];
  const float* W0 = (const float*)d_in[2];
  const float* b0 = (const float*)d_in[3];
  const float* W1 = (const float*)d_in[4];
  const float* b1 = (const float*)d_in[5];
  const float* W2 = (const float*)d_in[6];
  const float* b2 = (const float*)d_in[7];

  const int N = in_sizes[0] / 128;   // 100000
  const int E = in_sizes[1] / 2;     // 3200000
  const int* src = eidx;
  const int* dst = eidx + E;

  // workspace (floats): dinv[N] | hbuf[N*32] | agg[N*32] | lp[N] | agg2[N] | Wsw[4096]
  float* ws   = (float*)d_ws;
  float* dinv = ws;
  float* hbuf = dinv + N;
  float* agg  = hbuf + (size_t)N * 32;
  float* lp   = agg  + (size_t)N * 32;
  float* agg2 = lp + N;
  float* Wsw  = agg2 + N;

  const int BT = 256;
  const int gN   = (N + BT - 1) / BT;
  const int gE   = (E + BT - 1) / BT;
  const int gE8  = (E * 8 + BT - 1) / BT;        // 25.6M threads
  const int gN32 = (N * 32 + BT - 1) / BT;
  const int gTile = ((N + 15) / 16 + 3) / 4;     // 4 waves of 16 rows per block

  // normalization
  zero_f32<<<gN, BT, 0, stream>>>(dinv, N);
  deg_count<<<gE, BT, 0, stream>>>(dst, dinv, E);
  finalize_dinv<<<gN, BT, 0, stream>>>(dinv, N);

  // ---- layer 0: x @ W0 -> hbuf, aggregate, combine (+LeakyReLU)
  prepack_w<<<(128 * 32 + BT - 1) / BT, BT, 0, stream>>>(W0, Wsw, 128);
  gemm_wmma32<<<gTile, 128, 0, stream>>>(x, Wsw, hbuf, N, 128);
  zero_f32<<<gN32, BT, 0, stream>>>(agg, N * 32);
  scatter32<<<gE8, BT, 0, stream>>>(hbuf, src, dst, dinv, agg, E);
  combine32<<<gN32, BT, 0, stream>>>(hbuf, agg, dinv, b0, N);

  // ---- layer 1: hbuf @ W1 -> hbuf (in-place safe), aggregate, combine
  prepack_w<<<(32 * 32 + BT - 1) / BT, BT, 0, stream>>>(W1, Wsw, 32);
  gemm_wmma32<<<gTile, 128, 0, stream>>>(hbuf, Wsw, hbuf, N, 32);
  zero_f32<<<gN32, BT, 0, stream>>>(agg, N * 32);
  scatter32<<<gE8, BT, 0, stream>>>(hbuf, src, dst, dinv, agg, E);
  combine32<<<gN32, BT, 0, stream>>>(hbuf, agg, dinv, b1, N);

  // ---- layer 2: 32 -> 1 projection, aggregate, sigmoid head
  dot32<<<gN, BT, 0, stream>>>(hbuf, W2, lp, N);
  zero_f32<<<gN, BT, 0, stream>>>(agg2, N);
  scatter1<<<gE, BT, 0, stream>>>(lp, src, dst, dinv, agg2, E);
  final_head<<<gN, BT, 0, stream>>>(lp, agg2, dinv, b2, (float*)d_out, N);
}